// LGA3_55963423867525
// MI455X (gfx1250) — compile-verified
//
#include <hip/hip_runtime.h>
#include <stdint.h>

// Problem constants (match reference setup_inputs)
#define BB 2
#define DDIM 64
#define HH 384
#define WW 768
#define KK 5      // 2*RADIUS+1
#define NTAP 25   // KK*KK

// Tiling
#define TW 64
#define TH 4
#define DC 8                // disparity chunk per LDS stage (double buffered)
#define NCH (DDIM / DC)     // 8 chunks
#define ROWS (TH + 4)       // 8  (spatial halo +-2)
#define COLS (TW + 4)       // 68
#define SLICE (ROWS * COLS) // 544 floats per d-slice
#define NSL (DC + 2)        // 10 slices (d halo +-1)
#define BUF_ELEMS (NSL * SLICE) // 5440 floats = 21760 B per buffer

#define HW ((size_t)HH * WW)

// LDS byte offset of a pointer into a __shared__ array (addrspacecast to LDS,
// then ptrtoint -> 32-bit LDS offset; matches VDST semantics of async-to-LDS).
__device__ __forceinline__ unsigned lds_byte_off(const float* p) {
  return (unsigned)(uintptr_t)(__attribute__((address_space(3))) const float*)p;
}

// CDNA5 async copy: global -> LDS, 4 bytes per active lane, tracked by ASYNCcnt.
__device__ __forceinline__ void async_copy_f32(unsigned lds_off, const float* gaddr) {
  asm volatile("global_load_async_to_lds_b32 %0, %1, off"
               :: "v"(lds_off), "v"(gaddr)
               : "memory");
}

__device__ __forceinline__ void wait_async_all() {
  asm volatile("s_wait_asynccnt 0x0" ::: "memory");
}

__global__ __launch_bounds__(256)
void lga_pass_kernel(const float* __restrict__ src,   // [B, D, H, W]
                     const float* __restrict__ wgt,   // [B, 3, 25, H, W]
                     float* __restrict__ dst) {       // [B, D, H, W]
  __shared__ float smem[2][BUF_ELEMS];   // 43.5 KB double buffer

  const int tid = threadIdx.x;
  const int tx  = tid & (TW - 1);
  const int ty  = tid >> 6;
  const int w0  = blockIdx.x * TW;
  const int h0  = blockIdx.y * TH;
  const int b   = blockIdx.z;
  const int h   = h0 + ty;
  const int w   = w0 + tx;

  const float* sb = src + (size_t)b * DDIM * HW;

  // ---- Static per-thread staging slots (all scalars -> stay in VGPRs) ----
  // Slice = 544 floats, 256 threads:
  //   slot0: e0 = tid        (always < 544)
  //   slot1: e1 = tid + 256  (always < 512 < 544)
  //   slot2: e2 = tid + 512  (only for tid < 32)
  const int  e0 = tid;
  const int  e1 = tid + 256;
  const int  e2 = tid + 512;
  const int  r0 = e0 / COLS, q0 = e0 % COLS;
  const int  r1 = e1 / COLS, q1 = e1 % COLS;
  const int  r2 = e2 / COLS, q2 = e2 % COLS;
  const int  hs0 = h0 - 2 + r0, ws0 = w0 - 2 + q0;
  const int  hs1 = h0 - 2 + r1, ws1 = w0 - 2 + q1;
  const int  hs2 = h0 - 2 + r2, ws2 = w0 - 2 + q2;
  const int  soff0 = hs0 * WW + ws0;
  const int  soff1 = hs1 * WW + ws1;
  const int  soff2 = hs2 * WW + ws2;
  const bool sval0 = ((unsigned)hs0 < HH) && ((unsigned)ws0 < WW);
  const bool sval1 = ((unsigned)hs1 < HH) && ((unsigned)ws1 < WW);
  const bool sval2 = ((unsigned)hs2 < HH) && ((unsigned)ws2 < WW);
  const bool has2  = (tid < 32) && (e2 < SLICE);

  // ---- Preload this pixel's 75 weights into registers (reused over all D) ----
  float wr[3 * NTAP];
  {
    const float* wp = wgt + (size_t)b * 3 * NTAP * HW + (size_t)h * WW + w;
#pragma unroll
    for (int t = 0; t < 3 * NTAP; ++t)
      wr[t] = wp[(size_t)t * HW];
  }

  // ---- Stage one d-chunk into buf: slices d0-1 .. d0+DC, zero-padded ----
  auto stage = [&](float* buf, int d0) {
#pragma unroll 1
    for (int sl = 0; sl < NSL; ++sl) {
      const int  ds  = d0 - 1 + sl;                // global disparity
      const bool dok = (unsigned)ds < DDIM;
      const float* spb = sb + (size_t)ds * HW;
      float* lrow = buf + sl * SLICE;
      if (dok && sval0) async_copy_f32(lds_byte_off(lrow + e0), spb + soff0);
      else              lrow[e0] = 0.0f;
      if (dok && sval1) async_copy_f32(lds_byte_off(lrow + e1), spb + soff1);
      else              lrow[e1] = 0.0f;
      if (has2) {
        if (dok && sval2) async_copy_f32(lds_byte_off(lrow + e2), spb + soff2);
        else              lrow[e2] = 0.0f;
      }
    }
  };

  float acc[DC];

  // Prologue: stage chunk 0 into buffer 0.
  stage(&smem[0][0], 0);

  for (int c = 0; c < NCH; ++c) {
    const int d0 = c * DC;
    const float* buf = &smem[c & 1][0];

    wait_async_all();     // our async writes for this chunk have landed
    __syncthreads();      // everyone's have, and prior compute is done

    // Overlap: kick off async staging of the next chunk into the other buffer.
    if (c + 1 < NCH) stage(&smem[(c + 1) & 1][0], d0 + DC);

#pragma unroll
    for (int dd = 0; dd < DC; ++dd) acc[dd] = 0.0f;

    // ---- 25 taps x (d-1, d, d+1), rolling registers over d ----
#pragma unroll
    for (int i = 0; i < KK; ++i) {
#pragma unroll
      for (int j = 0; j < KK; ++j) {
        const int t = i * KK + j;
        const float g0 = wr[t];               // multiplies cost[d-1]
        const float g1 = wr[NTAP + t];        // multiplies cost[d]
        const float g2 = wr[2 * NTAP + t];    // multiplies cost[d+1]
        const float* sp = &buf[(ty + i) * COLS + (tx + j)];
        float c0 = sp[0];          // slice 0 -> d = d0-1
        float c1 = sp[SLICE];      // slice 1 -> d = d0
#pragma unroll
        for (int dd = 0; dd < DC; ++dd) {
          const float c2 = sp[(dd + 2) * SLICE];
          acc[dd] = fmaf(g0, c0, fmaf(g1, c1, fmaf(g2, c2, acc[dd])));
          c0 = c1;
          c1 = c2;
        }
      }
    }

    // ---- Coalesced stores ----
    float* op = dst + ((size_t)b * DDIM + d0) * HW + (size_t)h * WW + w;
#pragma unroll
    for (int dd = 0; dd < DC; ++dd)
      op[(size_t)dd * HW] = acc[dd];
  }
}

extern "C" void kernel_launch(void* const* d_in, const int* in_sizes, int n_in,
                              void* d_out, int out_size, void* d_ws, size_t ws_size,
                              hipStream_t stream) {
  const float* cost = (const float*)d_in[0];   // [2, 64, 384, 768] f32
  const float* wgt  = (const float*)d_in[1];   // [2, 75, 384, 768] f32
  float* out = (float*)d_out;
  float* tmp = (float*)d_ws;                   // needs >= 2*64*384*768*4 B

  dim3 grid(WW / TW, HH / TH, BB);
  dim3 block(256);

  // pass 1: in  -> out
  lga_pass_kernel<<<grid, block, 0, stream>>>(cost, wgt, out);
  // pass 2: out -> ws
  lga_pass_kernel<<<grid, block, 0, stream>>>(out, wgt, tmp);
  // pass 3: ws  -> out
  lga_pass_kernel<<<grid, block, 0, stream>>>(tmp, wgt, out);
}